// RGCNEncoder_19533511262868
// MI455X (gfx1250) — compile-verified
//
#include <hip/hip_runtime.h>

// RGCN encoder (2 layers) for MI455X / gfx1250.
// Edge phase: wave-per-edge block-diag transform, L2-resident f32 atomics.
// Root GEMM: fp32 WMMA (v_wmma_f32_16x16x4_f32) with K/col padding so the
// inner loop is guard-free (2x global_load_b64 + 1x v_wmma per K-step) and
// the epilogue is a clause of 8 agg loads + 8 stores at immediate offsets.

#define H_DIM 150
#define R_REL 35
#define NBLK  5
#define IBLK  30
#define LD    152   // padded row stride (multiple of 4, >= H_DIM)
#define NT16  160   // padded column count (10 tiles of 16)

typedef float v2f __attribute__((ext_vector_type(2)));
typedef float v8f __attribute__((ext_vector_type(8)));

__global__ void zero_f32(float* __restrict__ p, long n) {
  long i = (long)blockIdx.x * blockDim.x + threadIdx.x;
  long stride = (long)gridDim.x * blockDim.x;
  for (; i < n; i += stride) p[i] = 0.0f;
}

// xp[n][c] = c < 150 ? x[n][c] : 0   (row stride LD)
__global__ void pad_copy_kernel(const float* __restrict__ x,
                                float* __restrict__ xp, long total) {
  long i = (long)blockIdx.x * blockDim.x + threadIdx.x;
  if (i >= total) return;
  long n = i / LD;
  int c = (int)(i - n * LD);
  xp[i] = (c < H_DIM) ? x[n * H_DIM + c] : 0.0f;
}

// rootT[c][k] = (c<150 && k<150) ? root[k][c] : 0   (shape [NT16][LD])
__global__ void transpose_pad_kernel(const float* __restrict__ root,
                                     float* __restrict__ rootT) {
  int i = blockIdx.x * blockDim.x + threadIdx.x;
  if (i >= NT16 * LD) return;
  int c = i / LD;
  int k = i - c * LD;
  rootT[i] = (c < H_DIM && k < H_DIM) ? root[k * H_DIM + c] : 0.0f;
}

__global__ void count_kernel(const int* __restrict__ dst,
                             const int* __restrict__ et,
                             float* __restrict__ cnt, int E) {
  int i = blockIdx.x * blockDim.x + threadIdx.x;
  if (i < E) atomicAdd(&cnt[(long)dst[i] * R_REL + et[i]], 1.0f);
}

__global__ void inv_kernel(float* __restrict__ cnt, int n) {
  int i = blockIdx.x * blockDim.x + threadIdx.x;
  if (i < n) {
    float c = cnt[i];
    cnt[i] = (c > 0.0f) ? (1.0f / c) : 0.0f;
  }
}

// One wave32 per edge: gather x[src] (150 f32 -> 5 VGPRs, clamped loads, no
// EXEC masking), broadcast via __shfl, block-diag 5x(30x30) transform,
// scaled atomic scatter to agg[dst]. x/agg rows use stride LD.
__global__ __launch_bounds__(256) void edge_kernel(
    const float* __restrict__ x,        // [N,LD] padded
    const float* __restrict__ w,        // [R,B,IB,IB]
    const float* __restrict__ inv_cnt,  // [N*R]
    const int* __restrict__ src,
    const int* __restrict__ dst,
    const int* __restrict__ et,
    float* __restrict__ agg,            // [N,LD] accumulated
    int E) {
  int lane = threadIdx.x & 31;
  int wave = (int)(((long)blockIdx.x * blockDim.x + threadIdx.x) >> 5);
  if (wave >= E) return;  // wave-uniform

  int s = src[wave];
  int d = dst[wave];
  int r = et[wave];
  float inv = inv_cnt[(long)d * R_REL + r];

  const float* __restrict__ xr = x + (long)s * LD;
  float u[5];
#pragma unroll
  for (int k = 0; k < 5; ++k) {
    int idx = k * 32 + lane;
    idx = (idx < H_DIM) ? idx : 0;  // clamp: unconditional load, dup harmless
    u[k] = xr[idx];
  }

  const float* __restrict__ Wr = w + (long)r * (NBLK * IBLK * IBLK);
  int j = (lane < IBLK) ? lane : 0;  // lanes 30/31 compute garbage, never stored

#pragma unroll
  for (int b = 0; b < NBLK; ++b) {
    const float* __restrict__ Wb = Wr + b * IBLK * IBLK;
    float acc = 0.0f;
#pragma unroll
    for (int i = 0; i < IBLK; ++i) {
      const int t = b * IBLK + i;                 // compile-time after unroll
      float ui = __shfl(u[t >> 5], t & 31, 32);   // wave32 broadcast
      acc += ui * Wb[i * IBLK + j];
    }
    if (lane < IBLK)
      atomicAdd(&agg[(long)d * LD + b * IBLK + j], inv * acc);
  }
}

// One wave per 16x16 tile of out = act(agg + x @ root + bias).
// K runs over the padded LD=152 (38 x v_wmma_f32_16x16x4_f32), guard-free:
// A 16x4: lane%16 = M ; VGPR0 K = (lane<16?0:2), VGPR1 = +1 -> b64 load.
// B 4x16: lane%16 = N ; same K striping over transposed root -> b64 load.
// C/D:    lane%16 = N ; VGPR v holds M = v + (lane<16?0:8).
template <int DO_RELU>
__global__ __launch_bounds__(256) void root_gemm_kernel(
    const float* __restrict__ xp,     // [N,LD] padded
    const float* __restrict__ rootT,  // [NT16,LD] transposed+padded
    const float* __restrict__ agg,    // [N,LD]
    const float* __restrict__ bias,   // [H]
    float* __restrict__ out,          // [N,out_ld]
    int n_nodes, int out_ld) {
  const int ntiles = NT16 >> 4;  // 10
  int lane = threadIdx.x & 31;
  int gwave = (int)(((long)blockIdx.x * blockDim.x + threadIdx.x) >> 5);
  int mtile = gwave / ntiles;
  int ntile = gwave - mtile * ntiles;
  int m0 = mtile << 4;
  int n0 = ntile << 4;
  if (m0 >= n_nodes) return;  // wave-uniform: EXEC all-1s for WMMA

  int half = lane >> 4;  // 0: K pair {0,1}; 1: K pair {2,3}
  int l16  = lane & 15;
  int koff = half * 2;
  int m = m0 + l16;
  int mc = (m < n_nodes) ? m : (n_nodes - 1);  // clamp (N%16==0 anyway)
  int col = n0 + l16;                           // < NT16, always valid in rootT

  const float* __restrict__ ap = xp    + (long)mc  * LD + koff;
  const float* __restrict__ bp = rootT + (long)col * LD + koff;

  v8f c = {};
#pragma unroll 2
  for (int k0 = 0; k0 < LD; k0 += 4) {
    v2f a = *(const v2f*)(ap + k0);   // x[mc][k0+koff : +1], 8B aligned
    v2f b = *(const v2f*)(bp + k0);   // rootT[col][k0+koff : +1]
    c = __builtin_amdgcn_wmma_f32_16x16x4_f32(
        /*neg_a=*/false, a, /*neg_b=*/false, b,
        /*c_mod=*/(short)0, c, /*reuse_a=*/false, /*reuse_b=*/false);
  }

  if (col < out_ld) {
    int colc = (col < H_DIM) ? col : 0;
    float bb = bias[colc];
    bool colv = (col < H_DIM);
    int mbase = m0 + half * 8;
    const float* __restrict__ aggp = agg + (long)mbase * LD + col;
    float* __restrict__ outp = out + (long)mbase * out_ld + col;

    if (mbase + 8 <= n_nodes) {
      // Fast path (always taken for N%16==0): batched loads, immediate offsets.
      float av[8];
#pragma unroll
      for (int v = 0; v < 8; ++v) av[v] = aggp[v * LD];
#pragma unroll
      for (int v = 0; v < 8; ++v) {
        float val = c[v] + av[v] + bb;
        if (DO_RELU) val = fmaxf(val, 0.0f);
        if (!colv) val = 0.0f;  // zero the pad columns of h1
        outp[(long)v * out_ld] = val;
      }
    } else {
      for (int v = 0; v < 8; ++v) {
        if (mbase + v < n_nodes) {
          float val = c[v] + aggp[v * LD] + bb;
          if (DO_RELU) val = fmaxf(val, 0.0f);
          if (!colv) val = 0.0f;
          outp[(long)v * out_ld] = val;
        }
      }
    }
  }
}

extern "C" void kernel_launch(void* const* d_in, const int* in_sizes, int n_in,
                              void* d_out, int out_size, void* d_ws, size_t ws_size,
                              hipStream_t stream) {
  const float* node_emb = (const float*)d_in[0];
  const float* w1       = (const float*)d_in[1];
  const float* root1    = (const float*)d_in[2];
  const float* b1       = (const float*)d_in[3];
  const float* w2       = (const float*)d_in[4];
  const float* root2    = (const float*)d_in[5];
  const float* b2       = (const float*)d_in[6];
  const int*   eidx     = (const int*)d_in[7];  // [2,E]: src row then dst row
  const int*   etype    = (const int*)d_in[8];  // [E]

  const int N = in_sizes[0] / H_DIM;  // 20000
  const int E = in_sizes[8];          // 640000
  const int* src = eidx;
  const int* dst = eidx + E;

  // Workspace: cnt [N*R] | agg [N*LD] | h1 [N*LD] | x0 [N*LD] | rootT1 | rootT2
  float* cnt    = (float*)d_ws;
  float* agg    = cnt + (size_t)N * R_REL;
  float* h1     = agg + (size_t)N * LD;
  float* x0     = h1  + (size_t)N * LD;
  float* rootT1 = x0  + (size_t)N * LD;
  float* rootT2 = rootT1 + (size_t)NT16 * LD;
  float* out = (float*)d_out;
  (void)ws_size; (void)n_in; (void)out_size;

  const long padN = (long)N * LD;
  const int  edgeBlocks = (E + 7) / 8;  // 8 waves / 256-thread block
  const int  mt = (N + 15) / 16, nt = NT16 / 16;
  const long gthreads = (long)mt * nt * 32;
  const int  gemmBlocks = (int)((gthreads + 255) / 256);
  const int  tpBlocks = (NT16 * LD + 255) / 256;
  const int  pcBlocks = (int)((padN + 255) / 256);

  // Shared preprocessing: counts->reciprocals, padded x0, transposed roots.
  zero_f32<<<2048, 256, 0, stream>>>(cnt, (long)N * R_REL + padN);  // cnt+agg
  count_kernel<<<(E + 255) / 256, 256, 0, stream>>>(dst, etype, cnt, E);
  inv_kernel<<<((N * R_REL) + 255) / 256, 256, 0, stream>>>(cnt, N * R_REL);
  pad_copy_kernel<<<pcBlocks, 256, 0, stream>>>(node_emb, x0, padN);
  transpose_pad_kernel<<<tpBlocks, 256, 0, stream>>>(root1, rootT1);
  transpose_pad_kernel<<<tpBlocks, 256, 0, stream>>>(root2, rootT2);

  // Layer 1: agg = mean-scatter(W1 blockdiag . x0[src]); h1 = relu(agg + x0@root1 + b1)
  edge_kernel<<<edgeBlocks, 256, 0, stream>>>(x0, w1, cnt, src, dst, etype, agg, E);
  root_gemm_kernel<1><<<gemmBlocks, 256, 0, stream>>>(x0, rootT1, agg, b1, h1, N, LD);

  // Layer 2 on padded h1; final output unpadded [N,150].
  zero_f32<<<2048, 256, 0, stream>>>(agg, padN);
  edge_kernel<<<edgeBlocks, 256, 0, stream>>>(h1, w2, cnt, src, dst, etype, agg, E);
  root_gemm_kernel<0><<<gemmBlocks, 256, 0, stream>>>(h1, rootT2, agg, b2, out, N, H_DIM);
}